// NGCF_77850577207745
// MI455X (gfx1250) — compile-verified
//
#include <hip/hip_runtime.h>
#include <hip/hip_bf16.h>

#define DIM 128

typedef __attribute__((ext_vector_type(2))) float v2f;
typedef __attribute__((ext_vector_type(8))) float v8f;

// ---------------- zero workspace buffer ----------------
__global__ void ngcf_zero(float* __restrict__ p, long n) {
    long i = (long)blockIdx.x * blockDim.x + threadIdx.x;
    if (i < n) p[i] = 0.0f;
}

// ---------------- weighted edge scatter: msg[row] += w * ego[col] ----------------
// one wave (32 lanes) per edge; each lane handles 4 contiguous floats (float4)
__global__ void ngcf_scatter(const int* __restrict__ ei, const float* __restrict__ ew,
                             const float* __restrict__ ego, float* __restrict__ msg,
                             int E_) {
    int gid  = blockIdx.x * blockDim.x + threadIdx.x;
    int e    = gid >> 5;
    if (e >= E_) return;
    int lane = gid & 31;
    int row  = ei[e];
    int col  = ei[E_ + e];
    float w  = ew[e];
    float4 v = ((const float4*)(ego + (size_t)col * DIM))[lane];
    float* dst = msg + (size_t)row * DIM + lane * 4;
    atomicAdd(dst + 0, w * v.x);
    atomicAdd(dst + 1, w * v.y);
    atomicAdd(dst + 2, w * v.z);
    atomicAdd(dst + 3, w * v.w);
}

// ---------------- fused layer: leaky_relu(msg@gcW^T + gcb + (ego*msg)@biW^T + bib) ----------------
// 16 node rows per workgroup; 8 waves, each owns a 16-wide output-feature tile.
// f32 WMMA 16x16x4: exact fp32 matmul on the matrix pipe.
__global__ void __launch_bounds__(256)
ngcf_layer(const float* __restrict__ ego, const float* __restrict__ msg,
           const float* __restrict__ gcw, const float* __restrict__ gcb,
           const float* __restrict__ biw, const float* __restrict__ bib,
           float* __restrict__ ego_out) {
    const int node_base = blockIdx.x * 16;
    const int lane = threadIdx.x & 31;
    const int wave = threadIdx.x >> 5;   // 0..7 -> output feature tile
    const int nbase = wave * 16;
    const int half  = lane >> 4;         // 0/1 selects k-pair
    const int l15   = lane & 15;

    // A frags: row = node (l15), k = 2*half + j
    const float* msg_row = msg + (size_t)(node_base + l15) * DIM;
    const float* ego_row = ego + (size_t)(node_base + l15) * DIM;
    // B frags: B[k,n] = W[nbase+l15, k]  (x @ W^T)
    const float* gcw_row = gcw + (size_t)(nbase + l15) * DIM;
    const float* biw_row = biw + (size_t)(nbase + l15) * DIM;

    v8f acc_gc = {0.f,0.f,0.f,0.f,0.f,0.f,0.f,0.f};
    v8f acc_bi = {0.f,0.f,0.f,0.f,0.f,0.f,0.f,0.f};

    #pragma unroll 4
    for (int k = 0; k < DIM; k += 4) {
        const int kk = k + 2 * half;     // even -> 8B aligned float2
        v2f a_msg = *(const v2f*)(msg_row + kk);
        v2f a_ego = *(const v2f*)(ego_row + kk);
        v2f a_bi  = a_msg * a_ego;       // elementwise: frag layouts match
        v2f b_gc  = *(const v2f*)(gcw_row + kk);
        v2f b_bi  = *(const v2f*)(biw_row + kk);
        acc_gc = __builtin_amdgcn_wmma_f32_16x16x4_f32(false, a_msg, false, b_gc,
                                                       (short)0, acc_gc, false, false);
        acc_bi = __builtin_amdgcn_wmma_f32_16x16x4_f32(false, a_bi,  false, b_bi,
                                                       (short)0, acc_bi, false, false);
    }

    // D frag: lane l, VGPR j -> row j + 8*half, col l15
    const float bias = gcb[nbase + l15] + bib[nbase + l15];
    #pragma unroll
    for (int j = 0; j < 8; ++j) {
        int m = node_base + j + 8 * half;
        float x = acc_gc[j] + acc_bi[j] + bias;
        x = (x > 0.0f) ? x : 0.2f * x;   // leaky_relu, slope 0.2
        ego_out[(size_t)m * DIM + nbase + l15] = x;
    }
}

// ---------------- per-row inverse L2 norm ----------------
__global__ void ngcf_norm(const float* __restrict__ ego, float* __restrict__ invn, int N_) {
    int gid  = blockIdx.x * blockDim.x + threadIdx.x;
    int node = gid >> 5;
    if (node >= N_) return;
    int lane = gid & 31;
    float4 v = ((const float4*)(ego + (size_t)node * DIM))[lane];
    float ss = v.x*v.x + v.y*v.y + v.z*v.z + v.w*v.w;
    #pragma unroll
    for (int off = 16; off > 0; off >>= 1)
        ss += __shfl_xor(ss, off, 32);
    if (lane == 0) invn[node] = 1.0f / fmaxf(sqrtf(ss), 1e-12f);
}

// ---------------- pair scoring: sum over concat([emb, norm1, norm2, norm3]) ----------------
__global__ void ngcf_score(const int* __restrict__ eli, const float* __restrict__ emb,
                           const float* __restrict__ e1, const float* __restrict__ e2,
                           const float* __restrict__ e3, const float* __restrict__ invn,
                           float* __restrict__ out, int N_, int Q_) {
    int gid = blockIdx.x * blockDim.x + threadIdx.x;
    int q   = gid >> 5;
    if (q >= Q_) return;
    int lane = gid & 31;
    int s = eli[q];
    int t = eli[Q_ + q];

    float4 a = ((const float4*)(emb + (size_t)s * DIM))[lane];
    float4 b = ((const float4*)(emb + (size_t)t * DIM))[lane];
    float acc = a.x*b.x + a.y*b.y + a.z*b.z + a.w*b.w;

    const float* egos[3] = { e1, e2, e3 };
    #pragma unroll
    for (int i = 0; i < 3; ++i) {
        float4 u = ((const float4*)(egos[i] + (size_t)s * DIM))[lane];
        float4 w = ((const float4*)(egos[i] + (size_t)t * DIM))[lane];
        float d  = u.x*w.x + u.y*w.y + u.z*w.z + u.w*w.w;
        float sc = invn[(size_t)i * N_ + s] * invn[(size_t)i * N_ + t];
        acc += sc * d;   // (ego_s/|s|)·(ego_t/|t|) == (ego_s·ego_t)/(|s||t|)
    }
    #pragma unroll
    for (int off = 16; off > 0; off >>= 1)
        acc += __shfl_xor(acc, off, 32);
    if (lane == 0) out[q] = acc;
}

extern "C" void kernel_launch(void* const* d_in, const int* in_sizes, int n_in,
                              void* d_out, int out_size, void* d_ws, size_t ws_size,
                              hipStream_t stream) {
    const int*   ei  = (const int*)  d_in[0];   // [2,E]
    const int*   eli = (const int*)  d_in[1];   // [2,Q]
    const float* ew  = (const float*)d_in[2];   // [E]
    const float* emb = (const float*)d_in[3];   // [N,D]
    const float* gcw = (const float*)d_in[4];   // [L,D,D]
    const float* gcb = (const float*)d_in[5];   // [L,D]
    const float* biw = (const float*)d_in[6];   // [L,D,D]
    const float* bib = (const float*)d_in[7];   // [L,D]

    const int E_ = in_sizes[2];
    const int Q_ = in_sizes[1] / 2;
    const int N_ = in_sizes[3] / DIM;
    const int L_ = in_sizes[4] / (DIM * DIM);   // 3

    // workspace: msg [N*D] | ego1 [N*D] | ego2 [N*D] | ego3 [N*D] | invn [L*N]
    float* ws  = (float*)d_ws;
    const size_t nd = (size_t)N_ * DIM;
    float* msg  = ws;
    float* egoL[3] = { ws + nd, ws + 2 * nd, ws + 3 * nd };
    float* invn = ws + 4 * nd;

    const float* ego_cur = emb;
    for (int l = 0; l < L_; ++l) {
        ngcf_zero<<<(int)((nd + 255) / 256), 256, 0, stream>>>(msg, (long)nd);
        ngcf_scatter<<<(E_ * 32 + 255) / 256, 256, 0, stream>>>(ei, ew, ego_cur, msg, E_);
        ngcf_layer<<<N_ / 16, 256, 0, stream>>>(ego_cur, msg,
                                                gcw + (size_t)l * DIM * DIM, gcb + (size_t)l * DIM,
                                                biw + (size_t)l * DIM * DIM, bib + (size_t)l * DIM,
                                                egoL[l]);
        ngcf_norm<<<(N_ * 32 + 255) / 256, 256, 0, stream>>>(egoL[l], invn + (size_t)l * N_, N_);
        ego_cur = egoL[l];
    }
    ngcf_score<<<(Q_ * 32 + 255) / 256, 256, 0, stream>>>(eli, emb, egoL[0], egoL[1], egoL[2],
                                                          invn, (float*)d_out, N_, Q_);
}